// MeanGraphLayer_24584392802323
// MI455X (gfx1250) — compile-verified
//
#include <hip/hip_runtime.h>
#include <math.h>

#define N_NODES 100000
#define DIM     128
#define HID     512
#define NEDGE   600000

typedef float v2f __attribute__((ext_vector_type(2)));
typedef float v8f __attribute__((ext_vector_type(8)));

#define MB 32    // rows per workgroup
#define XS 260   // padded LDS stride for x block (256 cols)
#define HS 516   // padded LDS stride for hidden block (512 cols)

// ---------------- K0: zero scatter accumulators ----------------
__global__ void gnn_zero_kernel(float* __restrict__ agg, float* __restrict__ deg) {
    long long i = (long long)blockIdx.x * blockDim.x + threadIdx.x;
    if (i < (long long)N_NODES * DIM) agg[i] = 0.0f;
    if (i < N_NODES) deg[i] = 0.0f;
}

// ---------------- K1: edge-parallel scatter-sum (one wave32 per edge) ----------------
__global__ void gnn_scatter_kernel(const float* __restrict__ h,
                                   const long long* __restrict__ ei,
                                   float* __restrict__ agg,
                                   float* __restrict__ deg) {
    int e    = blockIdx.x * (blockDim.x >> 5) + (threadIdx.x >> 5);
    int lane = threadIdx.x & 31;
    if (e >= NEDGE) return;
    int src = (int)ei[e];
    int dst = (int)ei[NEDGE + e];
    const float4* hv = (const float4*)(h + (long long)src * DIM);
    float4 v = hv[lane];                       // global_load_b128, coalesced
    float* ap = agg + (long long)dst * DIM + lane * 4;
    unsafeAtomicAdd(ap + 0, v.x);              // global_atomic_add_f32
    unsafeAtomicAdd(ap + 1, v.y);
    unsafeAtomicAdd(ap + 2, v.z);
    unsafeAtomicAdd(ap + 3, v.w);
    if (lane == 0) unsafeAtomicAdd(deg + dst, 1.0f);
}

// ---------------- K2: fused mean-finalize + MLP via fp32 WMMA ----------------
__global__ __launch_bounds__(256)
void gnn_mlp_kernel(const float* __restrict__ h,
                    const float* agg,          // aliases `out` (row-disjoint phases)
                    const float* __restrict__ deg,
                    const float* __restrict__ W1,
                    const float* __restrict__ b1,
                    const float* __restrict__ W2,
                    const float* __restrict__ b2,
                    float* out) {
    extern __shared__ float smem[];
    float* xs = smem;            // [MB][XS] : x = [h | mean-agg]
    float* hs = smem + MB * XS;  // [MB][HS] : gelu(x@W1+b1)

    const int row0 = blockIdx.x * MB;
    const int tid  = threadIdx.x;
    const int lane = tid & 31;
    const int w    = tid >> 5;       // wave 0..7
    const int half = lane >> 4;      // 0/1 selects K{0,1} vs K{2,3}
    const int l15  = lane & 15;
    const int kOff = half * 2;

    // ---- stage x block: left half = h, right half = finalized scatter-mean ----
    for (int i = tid; i < MB * 2 * DIM; i += 256) {
        int r = i >> 8, c = i & 255;
        int gr = row0 + r;
        float v;
        if (c < DIM) {
            v = h[(long long)gr * DIM + c];
        } else {
            int c2 = c - DIM;
            float d = deg[gr];
            if (d > 0.5f) v = agg[(long long)gr * DIM + c2] / d;
            else          v = h[(long long)gr * DIM + c2];   // isolated node fallback
        }
        xs[r * XS + c] = v;
    }
    __syncthreads();

    // ---- GEMM1: hdn[MB][HID] = gelu(x @ W1 + b1); wave w owns cols [64w, 64w+64) ----
    {
        v8f zero = {};
        v8f acc[2][4];
        for (int mt = 0; mt < 2; ++mt)
            for (int nt = 0; nt < 4; ++nt) acc[mt][nt] = zero;
        const int nbase = w * 64;
        for (int kb = 0; kb < 2 * DIM; kb += 4) {
            v2f a[2];
            #pragma unroll
            for (int mt = 0; mt < 2; ++mt) {
                float2 t = *(const float2*)&xs[(mt * 16 + l15) * XS + kb + kOff]; // ds_load_b64
                a[mt].x = t.x; a[mt].y = t.y;
            }
            #pragma unroll
            for (int nt = 0; nt < 4; ++nt) {
                int n = nbase + nt * 16 + l15;
                v2f b;
                b.x = W1[(long long)(kb + kOff    ) * HID + n];
                b.y = W1[(long long)(kb + kOff + 1) * HID + n];
                #pragma unroll
                for (int mt = 0; mt < 2; ++mt)
                    acc[mt][nt] = __builtin_amdgcn_wmma_f32_16x16x4_f32(
                        false, a[mt], false, b, (short)0, acc[mt][nt], false, false);
            }
        }
        // epilogue: bias + exact-erf GELU, stash into LDS
        #pragma unroll
        for (int nt = 0; nt < 4; ++nt) {
            int col = nbase + nt * 16 + l15;
            float bias = b1[col];
            #pragma unroll
            for (int mt = 0; mt < 2; ++mt)
                #pragma unroll
                for (int r = 0; r < 8; ++r) {
                    float x = acc[mt][nt][r] + bias;
                    float g = 0.5f * x * (1.0f + erff(x * 0.70710678118654752f));
                    hs[(mt * 16 + half * 8 + r) * HS + col] = g;
                }
        }
    }
    __syncthreads();

    // ---- GEMM2: out[MB][DIM] = hdn @ W2 + b2; wave w owns cols [16w, 16w+16) ----
    {
        v8f zero = {};
        v8f acc[2] = {zero, zero};
        const int nb = w * 16;
        for (int kb = 0; kb < HID; kb += 4) {
            v2f b;
            b.x = W2[(long long)(kb + kOff    ) * DIM + nb + l15];
            b.y = W2[(long long)(kb + kOff + 1) * DIM + nb + l15];
            #pragma unroll
            for (int mt = 0; mt < 2; ++mt) {
                float2 t = *(const float2*)&hs[(mt * 16 + l15) * HS + kb + kOff]; // ds_load_b64
                v2f a; a.x = t.x; a.y = t.y;
                acc[mt] = __builtin_amdgcn_wmma_f32_16x16x4_f32(
                    false, a, false, b, (short)0, acc[mt], false, false);
            }
        }
        int col = nb + l15;
        float bias = b2[col];
        #pragma unroll
        for (int mt = 0; mt < 2; ++mt)
            #pragma unroll
            for (int r = 0; r < 8; ++r) {
                int rr = row0 + mt * 16 + half * 8 + r;
                out[(long long)rr * DIM + col] = acc[mt][r] + bias;
            }
    }
}

extern "C" void kernel_launch(void* const* d_in, const int* in_sizes, int n_in,
                              void* d_out, int out_size, void* d_ws, size_t ws_size,
                              hipStream_t stream) {
    const float*     h  = (const float*)d_in[0];
    const long long* ei = (const long long*)d_in[1];   // int64 edge_index [2,E]
    const float*     W1 = (const float*)d_in[2];       // [256,512]
    const float*     b1 = (const float*)d_in[3];
    const float*     W2 = (const float*)d_in[4];       // [512,128]
    const float*     b2 = (const float*)d_in[5];
    float* out = (float*)d_out;
    float* agg = out;                 // reuse output buffer as scatter accumulator
    float* deg = (float*)d_ws;        // N floats (400 KB) in workspace

    long long tot = (long long)N_NODES * DIM;
    int zb = (int)((tot + 255) / 256);
    gnn_zero_kernel<<<zb, 256, 0, stream>>>(agg, deg);

    int eb = (NEDGE + 7) / 8;         // 8 waves (edges) per 256-thread block
    gnn_scatter_kernel<<<eb, 256, 0, stream>>>(h, ei, agg, deg);

    int mb = N_NODES / MB;            // 3125, exact
    size_t shmem = (size_t)(MB * XS + MB * HS) * sizeof(float);  // ~99 KB
    gnn_mlp_kernel<<<mb, 256, shmem, stream>>>(h, agg, deg, W1, b1, W2, b2, out);
}